// Model_6571299963063
// MI455X (gfx1250) — compile-verified
//
#include <hip/hip_runtime.h>

#define N_NODES 50000
#define N_EDGES 800000
#define HIDDEN  128

typedef float v2f __attribute__((ext_vector_type(2)));
typedef float v8f __attribute__((ext_vector_type(8)));

// ---------------------------------------------------------------------------
// zero fill
// ---------------------------------------------------------------------------
__global__ void zero_kernel(float* __restrict__ p, unsigned long long n) {
    unsigned long long i = (unsigned long long)blockIdx.x * blockDim.x + threadIdx.x;
    if (i < n) p[i] = 0.0f;
}

// ---------------------------------------------------------------------------
// x0[i][:] = emb_table[node_id[i]][:]   (thread handles 4 floats)
// ---------------------------------------------------------------------------
__global__ void gather_kernel(const float* __restrict__ emb,
                              const int* __restrict__ node_id,
                              float* __restrict__ x0) {
    unsigned int t = blockIdx.x * blockDim.x + threadIdx.x;   // N*32 threads
    unsigned int node = t >> 5;
    unsigned int off  = (t & 31) * 4;
    if (node >= N_NODES) return;
    int srcn = node_id[node];
    const float4 v = *(const float4*)(emb + (unsigned long long)srcn * HIDDEN + off);
    *(float4*)(x0 + (unsigned long long)node * HIDDEN + off) = v;
}

// ---------------------------------------------------------------------------
// cnt[dst[e]] += 1.0f
// ---------------------------------------------------------------------------
__global__ void count_kernel(const int* __restrict__ dst, float* __restrict__ cnt) {
    unsigned int e = blockIdx.x * blockDim.x + threadIdx.x;
    if (e < N_EDGES) atomicAdd(&cnt[dst[e]], 1.0f);
}

// ---------------------------------------------------------------------------
// agg[dst[e]][:] += x[src[e]][:]   (one wave32 per edge, 4 floats per lane)
// ---------------------------------------------------------------------------
__global__ __launch_bounds__(256) void scatter_kernel(const float* __restrict__ x,
                                                      const int* __restrict__ src,
                                                      const int* __restrict__ dst,
                                                      float* __restrict__ agg) {
    unsigned int e    = (blockIdx.x * 256u + threadIdx.x) >> 5;
    unsigned int lane = threadIdx.x & 31;
    if (e >= N_EDGES) return;
    int s = src[e];
    int d = dst[e];
    const float4 v = *(const float4*)(x + (unsigned long long)s * HIDDEN + lane * 4);
    float* o = agg + (unsigned long long)d * HIDDEN + lane * 4;
    atomicAdd(o + 0, v.x);
    atomicAdd(o + 1, v.y);
    atomicAdd(o + 2, v.z);
    atomicAdd(o + 3, v.w);
}

// ---------------------------------------------------------------------------
// out = (agg / max(cnt,1)) @ Wl^T + bl + x @ Wr^T     [optionally ReLU]
//
// One block = 16 output rows x 128 output cols; wave w owns cols [16w,16w+16).
// f32 WMMA 16x16x4: 32 k-steps per 128-deep GEMM pass, two passes into one
// accumulator. Fragment addressing per CDNA5 ISA 7.12.2 32-bit layouts:
//   A: lane L -> row m0 + (L%16), K = 4*kk + 2*(L/16) + {0,1}   (b64 load)
//   B: lane L -> col n0 + (L%16), K = 4*kk + 2*(L/16) + {0,1}; B[k][n]=W[n][k]
//   C/D: lane L, vgpr v -> row m0 + v + 8*(L/16), col n0 + (L%16)
// ---------------------------------------------------------------------------
__global__ __launch_bounds__(256) void sage_gemm_kernel(const float* __restrict__ agg,
                                                        const float* __restrict__ cnt,
                                                        const float* __restrict__ x,
                                                        const float* __restrict__ Wl,
                                                        const float* __restrict__ bl,
                                                        const float* __restrict__ Wr,
                                                        float* __restrict__ out,
                                                        int relu) {
    const int m0   = blockIdx.x * 16;
    const int wave = threadIdx.x >> 5;      // 0..7
    const int lane = threadIdx.x & 31;
    const int n0   = wave * 16;
    const int r    = lane & 15;             // A-row / B-col within tile
    const int h    = lane >> 4;             // lane-half selects K pair
    const int koff = 2 * h;

    const int   mrow  = m0 + r;
    const float scale = 1.0f / fmaxf(cnt[mrow], 1.0f);

    const float* Arow  = agg + (unsigned long long)mrow * HIDDEN + koff;
    const float* Xrow  = x   + (unsigned long long)mrow * HIDDEN + koff;
    const float* WlRow = Wl  + (unsigned long long)(n0 + r) * HIDDEN + koff;
    const float* WrRow = Wr  + (unsigned long long)(n0 + r) * HIDDEN + koff;

    v8f c = {};

    // pass 1: mean-aggregated neighbors times Wl^T (scale folded into A frag)
    #pragma unroll
    for (int kk = 0; kk < 32; ++kk) {
        v2f a = *(const v2f*)(Arow + kk * 4);
        a.x *= scale;
        a.y *= scale;
        v2f b = *(const v2f*)(WlRow + kk * 4);
        c = __builtin_amdgcn_wmma_f32_16x16x4_f32(false, a, false, b,
                                                  (short)0, c, false, false);
    }
    // pass 2: self features times Wr^T
    #pragma unroll
    for (int kk = 0; kk < 32; ++kk) {
        v2f a = *(const v2f*)(Xrow + kk * 4);
        v2f b = *(const v2f*)(WrRow + kk * 4);
        c = __builtin_amdgcn_wmma_f32_16x16x4_f32(false, a, false, b,
                                                  (short)0, c, false, false);
    }

    const int   col  = n0 + r;
    const float bias = bl[col];
    #pragma unroll
    for (int v = 0; v < 8; ++v) {
        int   row = m0 + v + 8 * h;
        float o   = c[v] + bias;
        if (relu) o = fmaxf(o, 0.0f);
        out[(unsigned long long)row * HIDDEN + col] = o;
    }
}

// ---------------------------------------------------------------------------
// out[e] = dot(x[src[e]], x[dst[e]])   (one wave32 per edge)
// ---------------------------------------------------------------------------
__global__ __launch_bounds__(256) void edge_dot_kernel(const float* __restrict__ x,
                                                       const int* __restrict__ src,
                                                       const int* __restrict__ dst,
                                                       float* __restrict__ out) {
    unsigned int e    = (blockIdx.x * 256u + threadIdx.x) >> 5;
    unsigned int lane = threadIdx.x & 31;
    if (e >= N_EDGES) return;
    int s = src[e];
    int d = dst[e];
    const float4 a = *(const float4*)(x + (unsigned long long)s * HIDDEN + lane * 4);
    const float4 b = *(const float4*)(x + (unsigned long long)d * HIDDEN + lane * 4);
    float p = a.x * b.x + a.y * b.y + a.z * b.z + a.w * b.w;
    #pragma unroll
    for (int m = 16; m >= 1; m >>= 1) p += __shfl_xor(p, m, 32);
    if (lane == 0) out[e] = p;
}

// ---------------------------------------------------------------------------
// launcher
// ---------------------------------------------------------------------------
extern "C" void kernel_launch(void* const* d_in, const int* in_sizes, int n_in,
                              void* d_out, int out_size, void* d_ws, size_t ws_size,
                              hipStream_t stream) {
    const int*   node_id = (const int*)d_in[0];
    const int*   eidx    = (const int*)d_in[1];     // [2, E]
    const int*   src     = eidx;
    const int*   dst     = eidx + N_EDGES;
    const float* emb     = (const float*)d_in[2];
    const float* Wl1     = (const float*)d_in[3];
    const float* bl1     = (const float*)d_in[4];
    const float* Wr1     = (const float*)d_in[5];
    const float* Wl2     = (const float*)d_in[6];
    const float* bl2     = (const float*)d_in[7];
    const float* Wr2     = (const float*)d_in[8];
    float*       out     = (float*)d_out;

    const unsigned long long NH = (unsigned long long)N_NODES * HIDDEN;
    float* x0  = (float*)d_ws;            // layer-0 features (later reused for x2)
    float* agg = x0 + NH;                 // scatter accumulator
    float* x1  = agg + NH;                // layer-1 output
    float* cnt = x1 + NH;                 // degrees
    float* x2  = x0;                      // reuse: x0 dead after layer-1 GEMM

    const int ZB_NH  = (int)((NH + 255) / 256);
    const int ZB_N   = (N_NODES + 255) / 256;
    const int GATH   = (N_NODES * 32 + 255) / 256;
    const int CNTB   = (N_EDGES + 255) / 256;
    const int EDGEB  = (N_EDGES * 32 + 255) / 256;  // wave per edge
    const int GEMMB  = N_NODES / 16;                // 3125, exact

    zero_kernel<<<ZB_NH, 256, 0, stream>>>(agg, NH);
    zero_kernel<<<ZB_N, 256, 0, stream>>>(cnt, N_NODES);
    gather_kernel<<<GATH, 256, 0, stream>>>(emb, node_id, x0);
    count_kernel<<<CNTB, 256, 0, stream>>>(dst, cnt);

    // layer 1
    scatter_kernel<<<EDGEB, 256, 0, stream>>>(x0, src, dst, agg);
    sage_gemm_kernel<<<GEMMB, 256, 0, stream>>>(agg, cnt, x0, Wl1, bl1, Wr1, x1, 1);

    // layer 2
    zero_kernel<<<ZB_NH, 256, 0, stream>>>(agg, NH);
    scatter_kernel<<<EDGEB, 256, 0, stream>>>(x1, src, dst, agg);
    sage_gemm_kernel<<<GEMMB, 256, 0, stream>>>(agg, cnt, x1, Wl2, bl2, Wr2, x2, 0);

    // edge classifier
    edge_dot_kernel<<<EDGEB, 256, 0, stream>>>(x2, src, dst, out);
}